// HyperbolicAttention_25615184953662
// MI455X (gfx1250) — compile-verified
//
#include <hip/hip_runtime.h>
#include <math.h>

// ---------------------------------------------------------------------------
// Hyperbolic attention for MI455X (gfx1250, wave32, WMMA).
// All matmuls use V_WMMA_F32_16X16X4_F32 (fp32 in/out) because the hyperbolic
// distance arccosh(arg) is catastrophically sensitive near arg=1; fp16 inner
// products would wreck it, and at ~20 GFLOP total the f32 matrix pipe is ample.
// ---------------------------------------------------------------------------

#define BATCH 2
#define SEQ   2048
#define DMODEL 512
#define NHEAD 8
#define HDIM  64
#define EPS_F 1e-6f

typedef float v2f __attribute__((ext_vector_type(2)));
typedef float v8f __attribute__((ext_vector_type(8)));

__device__ __forceinline__ v8f wmma4(v2f a, v2f b, v8f c) {
  // D = A(16x4) * B(4x16) + C(16x16), fp32 throughout.
  return __builtin_amdgcn_wmma_f32_16x16x4_f32(
      /*neg_a=*/false, a, /*neg_b=*/false, b,
      /*c_mod=*/(short)0, c, /*reuse_a=*/false, /*reuse_b=*/false);
}

// ---------------------------------------------------------------------------
// Kernel 1: x_spatial = log_map_zero(x)[..., 1:]   (row norm over 512)
// ---------------------------------------------------------------------------
__global__ void prep_x(const float* __restrict__ x, float* __restrict__ xs) {
  __shared__ float red[128];
  const int row = blockIdx.x;          // b*L + l  (4096 rows)
  const int tid = threadIdx.x;         // 128 threads
  const float* xr = x + (size_t)row * (DMODEL + 1);
  float v[4]; float s = 0.0f;
  #pragma unroll
  for (int i = 0; i < 4; ++i) { v[i] = xr[1 + tid + i * 128]; s += v[i] * v[i]; }
  red[tid] = s; __syncthreads();
  for (int st = 64; st > 0; st >>= 1) {
    if (tid < st) red[tid] += red[tid + st];
    __syncthreads();
  }
  const float yn = fmaxf(sqrtf(red[0]), EPS_F);
  const float y0 = xr[0];
  const float d0 = acoshf(fminf(fmaxf(y0, 1.0f + EPS_F), 1e5f));
  const float f = d0 / yn;
  #pragma unroll
  for (int i = 0; i < 4; ++i)
    xs[(size_t)row * DMODEL + tid + i * 128] = v[i] * f;
}

// ---------------------------------------------------------------------------
// Kernel 2: WMMA GEMM  OUT[m, n] = sum_d A[m, d] * W[n, d]
//   M = 4096, N = 512, K = 512. Block = 256 thr (8 waves), tile 64x64, KC=32.
//   blockIdx.z selects (W, OUT).  mode 1 scatters into (B, H, L, 64) layout.
// ---------------------------------------------------------------------------
__global__ __launch_bounds__(256) void gemm_xw(
    const float* __restrict__ A,
    const float* __restrict__ W0, const float* __restrict__ W1,
    const float* __restrict__ W2,
    float* __restrict__ O0, float* __restrict__ O1, float* __restrict__ O2,
    int bhl_mode) {
  __shared__ float As[64][36];   // pad 36: float4-aligned, conflict-free
  __shared__ float Ws[64][36];
  const int tid = threadIdx.x;
  const int lane = tid & 31, wave = tid >> 5;
  const int mi16 = (wave & 3) * 16;        // 4 waves along M
  const int ni32 = (wave >> 2) * 32;       // 2 waves along N (2 tiles each)
  const int lr = lane & 15;
  const int kh = (lane >> 4) << 1;
  const int ro = (lane >> 4) << 3;
  const int m0 = blockIdx.y * 64, n0 = blockIdx.x * 64;

  const float* W = (blockIdx.z == 0) ? W0 : ((blockIdx.z == 1) ? W1 : W2);
  float* O       = (blockIdx.z == 0) ? O0 : ((blockIdx.z == 1) ? O1 : O2);

  v8f c0 = {}, c1 = {};
  for (int k0 = 0; k0 < DMODEL; k0 += 32) {
    __syncthreads();
    for (int i = tid; i < 512; i += 256) {      // 512 float4 per operand tile
      const int r = i >> 3, c4 = (i & 7) << 2;
      *(float4*)&As[r][c4] = *(const float4*)&A[(size_t)(m0 + r) * DMODEL + k0 + c4];
      *(float4*)&Ws[r][c4] = *(const float4*)&W[(size_t)(n0 + r) * DMODEL + k0 + c4];
    }
    if (k0 + 32 < DMODEL) {                     // gfx1250 global_prefetch_b8
      __builtin_prefetch(&A[(size_t)(m0 + (tid >> 2)) * DMODEL + k0 + 32], 0, 0);
      __builtin_prefetch(&W[(size_t)(n0 + (tid >> 2)) * DMODEL + k0 + 32], 0, 0);
    }
    __syncthreads();
    #pragma unroll
    for (int kk = 0; kk < 32; kk += 4) {
      v2f a  = *(const v2f*)&As[mi16 + lr][kk + kh];
      v2f b0 = *(const v2f*)&Ws[ni32 + lr][kk + kh];
      v2f b1 = *(const v2f*)&Ws[ni32 + 16 + lr][kk + kh];
      c0 = wmma4(a, b0, c0);
      c1 = wmma4(a, b1, c1);
    }
  }
  // epilogue
  #pragma unroll
  for (int i = 0; i < 8; ++i) {
    const int gr = m0 + mi16 + ro + i;
    const int col0 = n0 + ni32 + lr;
    const int col1 = col0 + 16;
    if (bhl_mode) {  // (B,H,L,64) scatter
      const int bb = gr >> 11, ll = gr & (SEQ - 1);
      const int h0 = col0 >> 6, h1 = col1 >> 6;
      O[(((size_t)bb * NHEAD + h0) * SEQ + ll) * HDIM + (col0 & 63)] = c0[i];
      O[(((size_t)bb * NHEAD + h1) * SEQ + ll) * HDIM + (col1 & 63)] = c1[i];
    } else {
      O[(size_t)gr * DMODEL + col0] = c0[i];
      O[(size_t)gr * DMODEL + col1] = c1[i];
    }
  }
}

// ---------------------------------------------------------------------------
// Kernel 3: in-place per-row (64-dim) maps.  One wave32 per row.
//  sel 0/1: q/k -> exp_map spatial (overwrite) + time component
//  sel 2  : v   -> log_map(exp_map(v)) roundtrip (overwrite)
// ---------------------------------------------------------------------------
__global__ __launch_bounds__(256) void map_qkv(
    float* __restrict__ q, float* __restrict__ k, float* __restrict__ v,
    float* __restrict__ qt, float* __restrict__ kt) {
  const int wave = threadIdx.x >> 5, lane = threadIdx.x & 31;
  const int row = blockIdx.x * 8 + wave;         // 0 .. B*H*L-1
  const int sel = blockIdx.y;
  float* base = (sel == 0) ? q : ((sel == 1) ? k : v);
  float2 val = *(float2*)&base[(size_t)row * HDIM + lane * 2];
  float ss = val.x * val.x + val.y * val.y;
  #pragma unroll
  for (int off = 16; off > 0; off >>= 1) ss += __shfl_xor(ss, off, 32);
  const float n = fmaxf(sqrtf(ss), EPS_F);       // clip(norm, min=EPS)
  const float scale = fminf(n, 15.0f);           // SQRT_K = 1
  const float sh = sinhf(scale);
  float f = sh / (n + EPS_F);
  if (sel == 2) {                                // log-map roundtrip factor
    const float tm = coshf(scale);
    const float d0 = acoshf(fminf(fmaxf(tm, 1.0f + EPS_F), 1e5f));
    const float yn = fmaxf(f * n, EPS_F);        // |spatial|
    f = f * d0 / yn;
  }
  val.x *= f; val.y *= f;
  *(float2*)&base[(size_t)row * HDIM + lane * 2] = val;
  if (lane == 0 && sel < 2) ((sel == 0) ? qt : kt)[row] = coshf(scale);
}

// ---------------------------------------------------------------------------
// Kernel 4: flash-style hyperbolic attention, 64x64 tiles, WMMA f32.
// grid = (L/64, B*H), block = 256 (8 waves).
// ---------------------------------------------------------------------------
__device__ __forceinline__ float score_fn(float space, float qtv, float ktv,
                                          float pq, int kidx, float wu) {
  const float arg = fminf(fmaxf(qtv * ktv - space, 1.0f + EPS_F), 1e5f);
  const float dist = acoshf(arg);
  float sc = -dist * dist - 0.1f * wu * log1pf(fabsf(pq - (float)kidx));
  return fminf(fmaxf(sc, -50.0f), 50.0f);
}

__global__ __launch_bounds__(256) void attn_kernel(
    const float* __restrict__ q, const float* __restrict__ k,
    const float* __restrict__ v, const float* __restrict__ qt,
    const float* __restrict__ kt, const float* __restrict__ mask,
    const int* __restrict__ pos_ids, const int* __restrict__ step,
    float* __restrict__ aout) {
  __shared__ float Qs[64][68];       // pad 68: float4-aligned, conflict-free
  __shared__ float KVs[64][68];      // holds K tile, then reused for V tile
  __shared__ float St[64][68];       // scores -> probs -> output staging
  __shared__ float qts[64], kts[64], posq[64];
  __shared__ float row_m[64], row_l[64], fac[64];

  const int tid = threadIdx.x;
  const int lane = tid & 31, wave = tid >> 5;
  const int mi16 = (wave & 3) * 16, ni32 = (wave >> 2) * 32;
  const int lr = lane & 15, kh = (lane >> 4) << 1, ro = (lane >> 4) << 3;
  const int qtile = blockIdx.x;
  const int bh = blockIdx.y;                 // b*H + h
  const int b = bh >> 3, h = bh & 7;
  const int q0 = qtile * 64;
  const size_t rb = (size_t)bh * SEQ * HDIM;
  const float wu = fminf(1.0f, (float)step[0] * (1.0f / 5000.0f));

  for (int i = tid; i < 1024; i += 256) {
    const int r = i >> 4, c4 = (i & 15) << 2;
    *(float4*)&Qs[r][c4] = *(const float4*)&q[rb + (size_t)(q0 + r) * HDIM + c4];
  }
  if (tid < 64) {
    qts[tid] = qt[(size_t)bh * SEQ + q0 + tid];
    posq[tid] = (float)pos_ids[(size_t)b * SEQ + q0 + tid];
    row_m[tid] = -INFINITY;
    row_l[tid] = 0.0f;
  }
  v8f o0 = {}, o1 = {};

  // Causal: mask is -1e9 beyond the diagonal -> softmax weight exactly 0,
  // so skipping k-tiles with k_start > q_end is exact. Diagonal tile is
  // handled by adding the mask itself.
  for (int kti = 0; kti <= qtile; ++kti) {
    const int k0 = kti * 64;
    __syncthreads();                           // prior P@V reads done
    for (int i = tid; i < 1024; i += 256) {
      const int r = i >> 4, c4 = (i & 15) << 2;
      *(float4*)&KVs[r][c4] = *(const float4*)&k[rb + (size_t)(k0 + r) * HDIM + c4];
    }
    if (tid < 64) kts[tid] = kt[(size_t)bh * SEQ + k0 + tid];
    __syncthreads();

    // ---- S = Qs * Ks^T (K-dim = 64 spatial dims) ----
    v8f s0 = {}, s1 = {};
    #pragma unroll
    for (int kk = 0; kk < HDIM; kk += 4) {
      v2f a  = *(const v2f*)&Qs[mi16 + lr][kk + kh];
      v2f b0 = *(const v2f*)&KVs[ni32 + lr][kk + kh];
      v2f b1 = *(const v2f*)&KVs[ni32 + 16 + lr][kk + kh];
      s0 = wmma4(a, b0, s0);
      s1 = wmma4(a, b1, s1);
    }
    // ---- hyperbolic score math, write to St ----
    #pragma unroll
    for (int i = 0; i < 8; ++i) {
      const int rl = mi16 + ro + i;
      const float qtv = qts[rl], pq = posq[rl];
      const int gq = q0 + rl;
      {
        const int cl = ni32 + lr;
        float sc = score_fn(s0[i], qtv, kts[cl], pq, k0 + cl, wu);
        sc += mask[((size_t)b * SEQ + gq) * SEQ + (k0 + cl)];
        St[rl][cl] = sc;
      }
      {
        const int cl = ni32 + 16 + lr;
        float sc = score_fn(s1[i], qtv, kts[cl], pq, k0 + cl, wu);
        sc += mask[((size_t)b * SEQ + gq) * SEQ + (k0 + cl)];
        St[rl][cl] = sc;
      }
    }
    __syncthreads();
    // ---- online softmax (threads 0..63) overlapped with V-tile load ----
    if (tid < 64) {
      const float mo = row_m[tid];
      float mt = -INFINITY;
      for (int c = 0; c < 64; ++c) mt = fmaxf(mt, St[tid][c]);
      const float mn = fmaxf(mo, mt);
      const float cr = __expf(mo - mn);
      float ls = 0.0f;
      for (int c = 0; c < 64; ++c) {
        const float p = __expf(St[tid][c] - mn);
        St[tid][c] = p;
        ls += p;
      }
      row_l[tid] = row_l[tid] * cr + ls;
      row_m[tid] = mn;
      fac[tid] = cr;
    } else {
      for (int i = tid - 64; i < 1024; i += 192) {
        const int r = i >> 4, c4 = (i & 15) << 2;
        *(float4*)&KVs[r][c4] =
            *(const float4*)&v[rb + (size_t)(k0 + r) * HDIM + c4];
      }
    }
    __syncthreads();
    // ---- rescale accumulators, O += P * Vt ----
    #pragma unroll
    for (int i = 0; i < 8; ++i) {
      const float cr = fac[mi16 + ro + i];
      o0[i] *= cr; o1[i] *= cr;
    }
    #pragma unroll
    for (int kk = 0; kk < 64; kk += 4) {
      v2f a = *(const v2f*)&St[mi16 + lr][kk + kh];
      v2f b0, b1;
      b0.x = KVs[kk + kh][ni32 + lr];      b0.y = KVs[kk + kh + 1][ni32 + lr];
      b1.x = KVs[kk + kh][ni32 + 16 + lr]; b1.y = KVs[kk + kh + 1][ni32 + 16 + lr];
      o0 = wmma4(a, b0, o0);
      o1 = wmma4(a, b1, o1);
    }
  }
  __syncthreads();
  // ---- normalize by row sum, stage O in LDS ----
  #pragma unroll
  for (int i = 0; i < 8; ++i) {
    const int rl = mi16 + ro + i;
    const float inv_l = 1.0f / row_l[rl];
    St[rl][ni32 + lr] = o0[i] * inv_l;
    St[rl][ni32 + 16 + lr] = o1[i] * inv_l;
  }
  __syncthreads();
  // ---- exp_map then log_map roundtrip factor per row ----
  if (tid < 64) {
    float ss = 0.0f;
    for (int c = 0; c < 64; ++c) { const float t = St[tid][c]; ss += t * t; }
    const float n = fmaxf(sqrtf(ss), EPS_F);
    const float scale = fminf(n, 15.0f);
    const float sh = sinhf(scale);
    const float f = sh / (n + EPS_F);
    const float tm = coshf(scale);
    const float d0 = acoshf(fminf(fmaxf(tm, 1.0f + EPS_F), 1e5f));
    const float yn = fmaxf(f * n, EPS_F);
    fac[tid] = f * d0 / yn;
  }
  __syncthreads();
  for (int i = tid; i < 4096; i += 256) {
    const int r = i >> 6, c = i & 63;
    aout[((size_t)b * SEQ + q0 + r) * DMODEL + h * HDIM + c] = St[r][c] * fac[r];
  }
}

// ---------------------------------------------------------------------------
// Kernel 5: final exp_map_zero over full 512-dim rows -> (B, L, 513)
// ---------------------------------------------------------------------------
__global__ void final_map(const float* __restrict__ ows, float* __restrict__ out) {
  __shared__ float red[128];
  const int row = blockIdx.x, tid = threadIdx.x;
  const float* r = ows + (size_t)row * DMODEL;
  float v[4]; float s = 0.0f;
  #pragma unroll
  for (int i = 0; i < 4; ++i) { v[i] = r[tid + i * 128]; s += v[i] * v[i]; }
  red[tid] = s; __syncthreads();
  for (int st = 64; st > 0; st >>= 1) {
    if (tid < st) red[tid] += red[tid + st];
    __syncthreads();
  }
  const float n = fmaxf(sqrtf(red[0]), EPS_F);
  const float scale = fminf(n, 15.0f);
  const float f = sinhf(scale) / (n + EPS_F);
  if (tid == 0) out[(size_t)row * (DMODEL + 1)] = coshf(scale);
  #pragma unroll
  for (int i = 0; i < 4; ++i)
    out[(size_t)row * (DMODEL + 1) + 1 + tid + i * 128] = v[i] * f;
}

// ---------------------------------------------------------------------------
extern "C" void kernel_launch(void* const* d_in, const int* in_sizes, int n_in,
                              void* d_out, int out_size, void* d_ws,
                              size_t ws_size, hipStream_t stream) {
  const float* x    = (const float*)d_in[0];
  const float* Wq   = (const float*)d_in[1];
  const float* Wk   = (const float*)d_in[2];
  const float* Wv   = (const float*)d_in[3];
  const float* Wo   = (const float*)d_in[4];
  const float* mask = (const float*)d_in[5];
  const int*   pos  = (const int*)d_in[6];   // JAX x64 off -> int32
  const int*   step = (const int*)d_in[7];

  float* ws = (float*)d_ws;
  const size_t NB = (size_t)BATCH * SEQ * DMODEL;  // 2,097,152 floats
  float* xs   = ws;            // (B*L, 512) log-mapped input
  float* qb   = ws + 1 * NB;   // (B,H,L,64) q proj -> qm spatial (in-place)
  float* kb   = ws + 2 * NB;
  float* vb   = ws + 3 * NB;   // -> v_tan (in-place)
  float* aout = ws + 4 * NB;   // (B,L,512) attention output (tangent)
  float* ows  = ws + 5 * NB;   // (B,L,512) after Wo
  float* qt   = ws + 6 * NB;   // (B,H,L) time components
  float* kt   = qt + (size_t)BATCH * NHEAD * SEQ;

  prep_x<<<BATCH * SEQ, 128, 0, stream>>>(x, xs);
  gemm_xw<<<dim3(DMODEL / 64, BATCH * SEQ / 64, 3), 256, 0, stream>>>(
      xs, Wq, Wk, Wv, qb, kb, vb, /*bhl_mode=*/1);
  map_qkv<<<dim3(BATCH * NHEAD * SEQ / 8, 3), 256, 0, stream>>>(qb, kb, vb, qt, kt);
  attn_kernel<<<dim3(SEQ / 64, BATCH * NHEAD), 256, 0, stream>>>(
      qb, kb, vb, qt, kt, mask, pos, step, aout);
  gemm_xw<<<dim3(DMODEL / 64, BATCH * SEQ / 64, 1), 256, 0, stream>>>(
      aout, Wo, Wo, Wo, ows, ows, ows, /*bhl_mode=*/0);
  final_map<<<BATCH * SEQ, 128, 0, stream>>>(ows, (float*)d_out);
}